// FloorPlaNet_52364241273518
// MI455X (gfx1250) — compile-verified
//
#include <hip/hip_runtime.h>
#include <hip/hip_bf16.h>

#define NBATCH 2048
#define NW 32
#define NP 6
#define WAVES 4
#define KDIM 12
#define INF_F 1.0e9f

typedef float v2f __attribute__((ext_vector_type(2)));
typedef float v8f __attribute__((ext_vector_type(8)));

// Per-wall geometric features exactly matching _param_distance's per-side terms.
__device__ __forceinline__ void wall_features(const float p[6], float feat[7], float& prob) {
    float sx = p[0], sy = p[1], ex = p[2], ey = p[3], w = p[4];
    prob = p[5];
    float difx = ex - sx, dify = ey - sy;
    float cx = 0.5f * (sx + ex), cy = 0.5f * (sy + ey);
    float len = sqrtf(difx * difx + dify * dify);
    float area = len * w;
    // side ratio(width, len)
    float smaller = fminf(w, len), bigger = fmaxf(w, len);
    float sr = (bigger > 0.f) ? (smaller / bigger) : 0.f;
    // bounding box
    float perx = dify, pery = -difx;
    if (len > 0.f) { perx /= len; pery /= len; }
    perx *= 0.5f * w; pery *= 0.5f * w;
    float right = fmaxf(fmaxf(sx + perx, sx - perx), fmaxf(ex + perx, ex - perx));
    float top   = fmaxf(fmaxf(sy + pery, sy - pery), fmaxf(ey + pery, ey - pery));
    float bbw = fabsf((right - cx) * 2.f);
    float bbh = fabsf((top  - cy) * 2.f);
    float bba = bbw * bbh;
    float ar = (bba > 0.001f) ? (area / bba) : 1.f;
    feat[0] = area; feat[1] = cx; feat[2] = cy; feat[3] = ar;
    feat[4] = bbw;  feat[5] = bbh; feat[6] = sr;
}

__global__ __launch_bounds__(WAVES * 32)
void floorplan_hungarian_kernel(const float* __restrict__ params_true,
                                const float* __restrict__ params_pred,
                                float* __restrict__ out)
{
    __shared__ float Af[WAVES][NW][KDIM];   // A rows (true side)
    __shared__ float Bf[WAVES][NW][KDIM];   // B cols (pred side)
    __shared__ float E[WAVES][NW * NW];     // 32x32 cost matrix per wave

    const int tid  = threadIdx.x;
    const int wave = tid >> 5;
    const int lane = tid & 31;
    const int b    = blockIdx.x * WAVES + wave;

    // ---- Phase 1: per-wall features (lane = wall index) ----
    {
        float p[6], feat[7], prob;
        const float* tp = params_true + ((size_t)b * NW + lane) * NP;
        #pragma unroll
        for (int k = 0; k < 6; k++) p[k] = tp[k];
        wall_features(p, feat, prob);
        float n2 = 0.f;
        #pragma unroll
        for (int k = 0; k < 7; k++) n2 += feat[k] * feat[k];
        float* a = Af[wave][lane];
        a[0] = 10.f * prob * prob + prob * n2;  // const-per-row term
        a[1] = 1.f;
        a[2] = prob;                            // pairs with -20*pp
        a[3] = prob;                            // pairs with |b|^2
        #pragma unroll
        for (int k = 0; k < 7; k++) a[4 + k] = prob * feat[k];
        a[11] = 0.f;

        const float* qp = params_pred + ((size_t)b * NW + lane) * NP;
        #pragma unroll
        for (int k = 0; k < 6; k++) p[k] = qp[k];
        wall_features(p, feat, prob);
        n2 = 0.f;
        #pragma unroll
        for (int k = 0; k < 7; k++) n2 += feat[k] * feat[k];
        float* bb = Bf[wave][lane];
        bb[0] = 1.f;
        bb[1] = 10.f * prob * prob;
        bb[2] = -20.f * prob;
        bb[3] = n2;
        #pragma unroll
        for (int k = 0; k < 7; k++) bb[4 + k] = -2.f * feat[k];
        bb[11] = 0.f;
    }
    __syncthreads();

    // ---- Phase 2: edges = A(32x12) x B(12x32) via V_WMMA_F32_16X16X4_F32 ----
    // 32-bit A 16x4 layout: lanes 0-15 -> K=0/1 in vgpr0/1, lanes 16-31 -> K=2/3.
    const int half = lane >> 4;
    const int mn   = lane & 15;
    #pragma unroll
    for (int rI = 0; rI < 2; rI++) {
        #pragma unroll
        for (int cJ = 0; cJ < 2; cJ++) {
            v8f acc = {};
            #pragma unroll
            for (int kc = 0; kc < 3; kc++) {
                const int kb = kc * 4 + half * 2;
                v2f a, bv;
                a.x  = Af[wave][rI * 16 + mn][kb];
                a.y  = Af[wave][rI * 16 + mn][kb + 1];
                bv.x = Bf[wave][cJ * 16 + mn][kb];
                bv.y = Bf[wave][cJ * 16 + mn][kb + 1];
                acc = __builtin_amdgcn_wmma_f32_16x16x4_f32(
                    false, a, false, bv, (short)0, acc, false, false);
            }
            // C/D layout: vgpr r, lanes 0-15 -> M=r, lanes 16-31 -> M=r+8, N=lane&15
            #pragma unroll
            for (int r = 0; r < 8; r++) {
                const int row = rI * 16 + r + half * 8;
                const int col = cJ * 16 + mn;
                E[wave][row * NW + col] = acc[r];
            }
        }
    }
    __syncthreads();

    // ---- Phase 3: Hungarian (shortest augmenting path), one wave per batch ----
    // lane l owns column l (v, spc, path, SC, row4col) and row l (u, col4row).
    const float* c = E[wave];
    float u_l = 0.f, v_l = 0.f;
    int col4row_l = -1, row4col_l = -1;

    for (int i0 = 0; i0 < NW; i0++) {
        int i = i0;
        unsigned SR = 0u;         // wave-uniform row mask
        bool  SC_l  = false;
        float spc_l = INF_F;
        int   path_l = -1;
        float minVal = 0.f;
        int   sink   = -1;
        while (sink < 0) {
            SR |= (1u << i);
            float u_i   = __shfl(u_l, i, 32);
            float slack = minVal + c[i * NW + lane] - u_i - v_l;
            if (!SC_l && slack < spc_l) { spc_l = slack; path_l = i; }
            float mv = SC_l ? INF_F : spc_l;
            int   mj = lane;
            #pragma unroll
            for (int off = 16; off > 0; off >>= 1) {  // argmin, first-index tiebreak
                float ov = __shfl_xor(mv, off, 32);
                int   oj = __shfl_xor(mj, off, 32);
                if (ov < mv || (ov == mv && oj < mj)) { mv = ov; mj = oj; }
            }
            minVal = mv;
            const int j = mj;
            if (lane == j) SC_l = true;
            const int r = __shfl(row4col_l, j, 32);
            sink = (r < 0) ? j : -1;
            i = r;
        }
        // dual update
        {
            float spc_c = __shfl(spc_l, col4row_l & 31, 32);  // spc[col4row[row]]
            bool in_sr = (SR >> lane) & 1u;
            if (in_sr && lane != i0) u_l += minVal - spc_c;
            if (lane == i0)          u_l += minVal;
            if (SC_l)                v_l -= (minVal - spc_l);
        }
        // augment along path
        int j = sink;
        while (true) {
            const int ii = __shfl(path_l, j, 32);
            if (lane == j)  row4col_l = ii;
            const int jn = __shfl(col4row_l, ii, 32);
            if (lane == ii) col4row_l = j;
            j = jn;
            if (ii == i0) break;
        }
    }

    // ---- Phase 4: loss = sum of matched edges ----
    float e = c[lane * NW + col4row_l];
    #pragma unroll
    for (int off = 16; off > 0; off >>= 1) e += __shfl_xor(e, off, 32);
    if (lane == 0) out[b] = e;
}

extern "C" void kernel_launch(void* const* d_in, const int* in_sizes, int n_in,
                              void* d_out, int out_size, void* d_ws, size_t ws_size,
                              hipStream_t stream) {
    const float* params_true = (const float*)d_in[0];
    const float* params_pred = (const float*)d_in[1];
    float* out = (float*)d_out;
    dim3 block(WAVES * 32);
    dim3 grid(NBATCH / WAVES);
    floorplan_hungarian_kernel<<<grid, block, 0, stream>>>(params_true, params_pred, out);
}